// SVDTruncation_32573031973057
// MI455X (gfx1250) — compile-verified
//
#include <hip/hip_runtime.h>

// Batched truncated SVD via subspace iteration on G = A^T A, reconstruct A*Q*Q^T.
// One 64x64 fp32 matrix per 128-thread (4-wave) workgroup; all GEMMs on
// v_wmma_f32_16x16x4_f32; A staged with async global->LDS copies when available.

typedef float v2f __attribute__((ext_vector_type(2)));
typedef float v8f __attribute__((ext_vector_type(8)));
typedef int v4i_vs __attribute__((vector_size(16)));  // matches builtin param type
typedef __attribute__((address_space(1))) v4i_vs* gptr_v4i;
typedef __attribute__((address_space(3))) v4i_vs* lptr_v4i;

#define LDA 65   // padded leading dim for 64-wide LDS arrays (64 banks)
#define LDQ 17   // padded leading dim for 16-wide LDS arrays
#define NITER 12 // subspace iteration count

#if defined(__has_builtin)
#if __has_builtin(__builtin_amdgcn_global_load_async_to_lds_b128)
#define HAVE_ASYNC_LDS 1
#endif
#if __has_builtin(__builtin_amdgcn_s_wait_asynccnt)
#define HAVE_WAIT_ASYNC 1
#endif
#endif

__device__ __forceinline__ v8f wmma4(v2f a, v2f b, v8f c) {
  // D = A(16x4,f32) * B(4x16,f32) + C(16x16,f32)
  return __builtin_amdgcn_wmma_f32_16x16x4_f32(
      /*neg_a=*/false, a, /*neg_b=*/false, b,
      /*c_mod=*/(short)0, c, /*reuse_a=*/false, /*reuse_b=*/false);
}

// A-fragment (16x4), row-major source: elem(m,k) = base[(m0+m)*ld + k0+k]
__device__ __forceinline__ v2f frag_a(const float* base, int ld, int m0, int k0, int lane) {
  int ml = lane & 15, kh = (lane >> 4) << 1;
  v2f a;
  a.x = base[(m0 + ml) * ld + k0 + kh + 0];
  a.y = base[(m0 + ml) * ld + k0 + kh + 1];
  return a;
}

// A-fragment of transposed source: elem(m,k) = base[(k0+k)*ld + m0+m]
__device__ __forceinline__ v2f frag_aT(const float* base, int ld, int m0, int k0, int lane) {
  int ml = lane & 15, kh = (lane >> 4) << 1;
  v2f a;
  a.x = base[(k0 + kh + 0) * ld + m0 + ml];
  a.y = base[(k0 + kh + 1) * ld + m0 + ml];
  return a;
}

// B-fragment (4x16), row-major source: elem(k,n) = base[(k0+k)*ld + n0+n]
__device__ __forceinline__ v2f frag_b(const float* base, int ld, int k0, int n0, int lane) {
  int nl = lane & 15, kh = (lane >> 4) << 1;
  v2f b;
  b.x = base[(k0 + kh + 0) * ld + n0 + nl];
  b.y = base[(k0 + kh + 1) * ld + n0 + nl];
  return b;
}

// B-fragment of transposed source: elem(k,n) = base[(n0+n)*ld + k0+k]
__device__ __forceinline__ v2f frag_bT(const float* base, int ld, int k0, int n0, int lane) {
  int nl = lane & 15, kh = (lane >> 4) << 1;
  v2f b;
  b.x = base[(n0 + nl) * ld + k0 + kh + 0];
  b.y = base[(n0 + nl) * ld + k0 + kh + 1];
  return b;
}

// C/D tile (16x16 f32): VGPR r, lanes 0-15 -> M=r, lanes 16-31 -> M=r+8; N = lane&15
__device__ __forceinline__ void store_tile(float* base, int ld, int m0, int n0, int lane, v8f c) {
  int nl = lane & 15, mh = (lane >> 4) << 3;
#pragma unroll
  for (int r = 0; r < 8; ++r) base[(m0 + mh + r) * ld + n0 + nl] = c[r];
}

__global__ __launch_bounds__(128)
void svd_trunc64_k16_kernel(const float* __restrict__ x, float* __restrict__ out) {
  __shared__ float lA[64 * LDA];  // input matrix
  __shared__ float lG[64 * LDA];  // G = A^T A
  __shared__ float lQ[64 * LDQ];  // orthonormal basis (64x16)
  __shared__ float lZ[64 * LDQ];  // Z = G*Q, then Y = A*Q
  __shared__ float lR[16 * LDQ];  // M = Z^T Z, then Cholesky factor R (upper)

  const int t = threadIdx.x;
  const int lane = t & 31;
  const int w = __builtin_amdgcn_readfirstlane(t >> 5);  // wave id 0..3, uniform
  const size_t blk = blockIdx.x;
  const float* Ag = x + blk * 4096;
  float* Og = out + blk * 4096;

  // ---- Stage A into LDS (4096 floats, padded rows of 65) ----
#ifdef HAVE_ASYNC_LDS
  // async global->LDS, 16B per lane per op, no VGPR round trip (ASYNCcnt path)
#pragma unroll
  for (int c = 0; c < 8; ++c) {
    int e = (c * 128 + t) * 4;  // element index, 4-float group stays in one row
    int row = e >> 6, col = e & 63;
    __builtin_amdgcn_global_load_async_to_lds_b128(
        (gptr_v4i)(Ag + e), (lptr_v4i)(lA + row * LDA + col),
        /*offset=*/0, /*cpol=*/0);
  }
#ifdef HAVE_WAIT_ASYNC
  __builtin_amdgcn_s_wait_asynccnt(0);
#else
  asm volatile("s_wait_asynccnt 0x0" ::: "memory");
#endif
#else
#pragma unroll
  for (int c = 0; c < 8; ++c) {
    int e = (c * 128 + t) * 4;
    float4 v = *reinterpret_cast<const float4*>(Ag + e);
    int row = e >> 6, col = e & 63;
    lA[row * LDA + col + 0] = v.x;
    lA[row * LDA + col + 1] = v.y;
    lA[row * LDA + col + 2] = v.z;
    lA[row * LDA + col + 3] = v.w;
  }
#endif
  __syncthreads();

  // ---- G = A^T A : 4x4 grid of 16x16 tiles, 4 tiles per wave (column w) ----
#pragma unroll
  for (int p = 0; p < 4; ++p) {
    int m0 = p * 16, n0 = w * 16;
    v8f c = {};
#pragma unroll
    for (int k0 = 0; k0 < 64; k0 += 4) {
      v2f a = frag_aT(lA, LDA, m0, k0, lane);  // (A^T)[m][k] = A[k][m]
      v2f b = frag_b(lA, LDA, k0, n0, lane);
      c = wmma4(a, b, c);
    }
    store_tile(lG, LDA, m0, n0, lane, c);
  }
  __syncthreads();

  // ---- Q0 = [I16 ; 0] ----
  for (int i = t; i < 64 * 16; i += 128) {
    int row = i >> 4, col = i & 15;
    lQ[row * LDQ + col] = (row == col) ? 1.0f : 0.0f;
  }
  __syncthreads();

  // ---- Subspace iteration with CholeskyQR re-orthonormalization ----
  for (int it = 0; it < NITER; ++it) {
    // Z = G * Q  (64x16): all 4 waves, one 16x16 row-tile each
    {
      int m0 = w * 16;
      v8f c = {};
#pragma unroll
      for (int k0 = 0; k0 < 64; k0 += 4) {
        v2f a = frag_a(lG, LDA, m0, k0, lane);
        v2f b = frag_b(lQ, LDQ, k0, 0, lane);
        c = wmma4(a, b, c);
      }
      store_tile(lZ, LDQ, m0, 0, lane, c);
    }
    __syncthreads();

    // M = Z^T Z  (16x16): wave 0, single tile accumulated over K=64
    if (w == 0) {
      v8f c = {};
#pragma unroll
      for (int k0 = 0; k0 < 64; k0 += 4) {
        v2f a = frag_aT(lZ, LDQ, 0, k0, lane);
        v2f b = frag_b(lZ, LDQ, k0, 0, lane);
        c = wmma4(a, b, c);
      }
      store_tile(lR, LDQ, 0, 0, lane, c);
    }
    __syncthreads();

    // Cholesky M = R^T R (upper triangular, in place in lR), single thread
    if (t == 0) {
      for (int j = 0; j < 16; ++j) {
        float d = lR[j * LDQ + j];
        for (int i = 0; i < j; ++i) {
          float rij = lR[i * LDQ + j];
          d -= rij * rij;
        }
        d = sqrtf(fmaxf(d, 1e-20f));
        lR[j * LDQ + j] = d;
        float inv = 1.0f / d;
        for (int k = j + 1; k < 16; ++k) {
          float s = lR[j * LDQ + k];
          for (int i = 0; i < j; ++i) s -= lR[i * LDQ + j] * lR[i * LDQ + k];
          lR[j * LDQ + k] = s * inv;
        }
      }
    }
    __syncthreads();

    // Q = Z * R^{-1}: one row per thread (threads 0..63), forward substitution
    if (t < 64) {
      float q[16];
#pragma unroll
      for (int j = 0; j < 16; ++j) {
        float s = lZ[t * LDQ + j];
        for (int i = 0; i < j; ++i) s -= q[i] * lR[i * LDQ + j];
        q[j] = s / lR[j * LDQ + j];
      }
#pragma unroll
      for (int j = 0; j < 16; ++j) lQ[t * LDQ + j] = q[j];
    }
    __syncthreads();
  }

  // ---- Y = A * Q  (64x16), reuse lZ: all 4 waves ----
  {
    int m0 = w * 16;
    v8f c = {};
#pragma unroll
    for (int k0 = 0; k0 < 64; k0 += 4) {
      v2f a = frag_a(lA, LDA, m0, k0, lane);
      v2f b = frag_b(lQ, LDQ, k0, 0, lane);
      c = wmma4(a, b, c);
    }
    store_tile(lZ, LDQ, m0, 0, lane, c);
  }
  __syncthreads();

  // ---- out = Y * Q^T  (64x64): 16 tiles, 4 per wave, K = 16; store to global ----
#pragma unroll
  for (int p = 0; p < 4; ++p) {
    int m0 = p * 16, n0 = w * 16;
    v8f c = {};
#pragma unroll
    for (int k0 = 0; k0 < 16; k0 += 4) {
      v2f a = frag_a(lZ, LDQ, m0, k0, lane);
      v2f b = frag_bT(lQ, LDQ, k0, n0, lane);  // (Q^T)[k][n] = Q[n][k]
      c = wmma4(a, b, c);
    }
    int nl = lane & 15, mh = (lane >> 4) << 3;
#pragma unroll
    for (int r = 0; r < 8; ++r) Og[(m0 + mh + r) * 64 + n0 + nl] = c[r];
  }
}

extern "C" void kernel_launch(void* const* d_in, const int* in_sizes, int n_in,
                              void* d_out, int out_size, void* d_ws, size_t ws_size,
                              hipStream_t stream) {
  const float* x = (const float*)d_in[0];
  float* out = (float*)d_out;
  int nblocks = in_sizes[0] / 4096;  // 16384 independent 64x64 matrices
  svd_trunc64_k16_kernel<<<nblocks, 128, 0, stream>>>(x, out);
}